// BoxModelStable_352187318794
// MI455X (gfx1250) — compile-verified
//
#include <hip/hip_runtime.h>

#define NUM_MODELS 2
#define NUM_BOXES 100000
#define DIMS 64
#define N_PAIRS 500000
#define BOX_STRIDE (2 * DIMS)                    // 128 floats per (model, box)
#define MODEL_STRIDE ((size_t)NUM_BOXES * BOX_STRIDE)
#define LN2F     0.69314718055994530942f
#define INV_LN2F 1.44269504088896340736f
// EPS = np.finfo(float32).tiny = smallest normal = bit pattern 0x00800000

typedef __attribute__((ext_vector_type(2))) float v2f;
typedef __attribute__((ext_vector_type(8))) float v8f;

// raw hardware transcendentals: inputs >= smallest normal, no denorm fixup needed
__device__ __forceinline__ float rlog2(float x) { return __builtin_amdgcn_logf(x); }   // v_log_f32
__device__ __forceinline__ float rexp2(float x) { return __builtin_amdgcn_exp2f(x); }  // v_exp_f32

// log2 of intersection width along one dim, exact 4-op chain via inline asm.
// clip01 is monotone => clip(max(a,b)) == max(clip(a),clip(b)), fused into the
// clamp destination modifier. Inline asm also avoids the v_max_num x,x NaN
// canonicalization LLVM inserts on values coming straight from loads.
__device__ __forceinline__ float iterm2(float aLo, float bLo, float aHi, float bHi) {
    float lo, hi, d, w;
    asm("v_max_num_f32 %0, %1, %2 clamp" : "=v"(lo) : "v"(aLo), "v"(bLo));
    asm("v_min_num_f32 %0, %1, %2 clamp" : "=v"(hi) : "v"(aHi), "v"(bHi));
    asm("v_sub_f32 %0, %1, %2"           : "=v"(d)  : "v"(hi),  "v"(lo));
    asm("v_max_num_f32 %0, 0x800000, %1" : "=v"(w)  : "v"(d));
    return rlog2(w);
}

// log2 width of a single box along one dim (clip01 both ends, one clamp op each)
__device__ __forceinline__ float uterm2(float loRaw, float hiRaw) {
    float lo, hi, d, w;
    asm("v_max_num_f32 %0, %1, %1 clamp" : "=v"(lo) : "v"(loRaw));
    asm("v_max_num_f32 %0, %1, %1 clamp" : "=v"(hi) : "v"(hiRaw));
    asm("v_sub_f32 %0, %1, %2"           : "=v"(d)  : "v"(hi), "v"(lo));
    asm("v_max_num_f32 %0, 0x800000, %1" : "=v"(w)  : "v"(d));
    return rlog2(w);
}

// natural-log logsumexp of two finite values, computed with base-2 hardware ops
__device__ __forceinline__ float lse2(float a, float b) {
    float m = fmaxf(a, b);
    float s = rexp2((a - m) * INV_LN2F) + rexp2((b - m) * INV_LN2F);
    return m + LN2F * rlog2(s);
}

// ---------------------------------------------------------------------------
// Kernel 1: log_unary_probs[box] = lse(w + log_vol(box)) - lse(w)
// One wave per box; each lane handles 2 dims in log2 domain; shfl_xor reduce.
// ---------------------------------------------------------------------------
__global__ __launch_bounds__(256) void unary_kernel(const float* __restrict__ emb,
                                                    const float* __restrict__ weights,
                                                    float* __restrict__ out_unary) {
    const int wave = threadIdx.x >> 5;
    const int lane = threadIdx.x & 31;
    const int box  = blockIdx.x * 8 + wave;
    if (box >= NUM_BOXES) return;

    float part[NUM_MODELS];
#pragma unroll
    for (int m = 0; m < NUM_MODELS; ++m) {
        const float* base = emb + (size_t)m * MODEL_STRIDE + (size_t)box * BOX_STRIDE;
        float2 lo = *(const float2*)(base + 2 * lane);
        float2 hi = *(const float2*)(base + DIMS + 2 * lane);
        part[m] = uterm2(lo.x, hi.x) + uterm2(lo.y, hi.y);
    }
#pragma unroll
    for (int off = 16; off >= 1; off >>= 1) {
        part[0] += __shfl_xor(part[0], off, 32);
        part[1] += __shfl_xor(part[1], off, 32);
    }
    if (lane == 0) {
        float w0 = weights[0], w1 = weights[1];
        out_unary[box] = lse2(w0 + LN2F * part[0], w1 + LN2F * part[1]) - lse2(w0, w1);
    }
}

// ---------------------------------------------------------------------------
// Kernel 2: per-pair intersection log-volume; the 64-dim reduction runs on the
// matrix pipe via V_WMMA_F32_16X16X4_F32 with B = ones. Because B is all-ones,
// K-order inside each 4-wide block is irrelevant, so each lane loads one
// float4 (b128) of 4 contiguous dims and feeds TWO chained WMMA steps.
// Both models interleave in one loop: two independent accumulator chains, so
// WMMA latency overlaps with the other chain's VALU/log work.
// Accumulators hold sums of log2 widths; scaled by ln2 once at finalize.
// ---------------------------------------------------------------------------
__global__ __launch_bounds__(256) void pair_kernel(const int* __restrict__ idx,
                                                   const float* __restrict__ emb,
                                                   const float* __restrict__ weights,
                                                   const float* __restrict__ unary,
                                                   float* __restrict__ out_logp,
                                                   float* __restrict__ out_p) {
    __shared__ float s[8][NUM_MODELS][16];

    const int wave  = threadIdx.x >> 5;
    const int lane  = threadIdx.x & 31;
    const int mrow  = lane & 15;          // M row this lane feeds
    const int half  = lane >> 4;          // 0 or 1: which 4-dim chunk per 8-dim step
    const int pbase = (blockIdx.x * 8 + wave) * 16;

    int p = pbase + mrow;
    if (p >= N_PAIRS) p = N_PAIRS - 1;    // clamp (no masking): EXEC all-ones for WMMA
    const int i0 = idx[2 * p + 0];
    const int i1 = idx[2 * p + 1];

    const v2f bones = {1.0f, 1.0f};       // B = ones(4x16): D[i][j] = sum_k A[i][k] + C

    const float* a0 = emb + (size_t)i0 * BOX_STRIDE;   // model 0, box A
    const float* b0 = emb + (size_t)i1 * BOX_STRIDE;   // model 0, box B
    const float* a1 = a0 + MODEL_STRIDE;               // model 1, box A
    const float* b1 = b0 + MODEL_STRIDE;               // model 1, box B

    v8f c0 = {};
    v8f c1 = {};
#pragma unroll
    for (int t = 0; t < 8; ++t) {
        const int k = t * 8 + half * 4;                // 16B-aligned float4
        float4 a0lo = *(const float4*)(a0 + k);
        float4 a0hi = *(const float4*)(a0 + DIMS + k);
        float4 b0lo = *(const float4*)(b0 + k);
        float4 b0hi = *(const float4*)(b0 + DIMS + k);
        float4 a1lo = *(const float4*)(a1 + k);
        float4 a1hi = *(const float4*)(a1 + DIMS + k);
        float4 b1lo = *(const float4*)(b1 + k);
        float4 b1hi = *(const float4*)(b1 + DIMS + k);

        v2f av;
        av[0] = iterm2(a0lo.x, b0lo.x, a0hi.x, b0hi.x);
        av[1] = iterm2(a0lo.y, b0lo.y, a0hi.y, b0hi.y);
        c0 = __builtin_amdgcn_wmma_f32_16x16x4_f32(false, av, false, bones,
                                                   (short)0, c0, false, false);
        av[0] = iterm2(a1lo.x, b1lo.x, a1hi.x, b1hi.x);
        av[1] = iterm2(a1lo.y, b1lo.y, a1hi.y, b1hi.y);
        c1 = __builtin_amdgcn_wmma_f32_16x16x4_f32(false, av, false, bones,
                                                   (short)0, c1, false, false);
        av[0] = iterm2(a0lo.z, b0lo.z, a0hi.z, b0hi.z);
        av[1] = iterm2(a0lo.w, b0lo.w, a0hi.w, b0hi.w);
        c0 = __builtin_amdgcn_wmma_f32_16x16x4_f32(false, av, false, bones,
                                                   (short)0, c0, false, false);
        av[0] = iterm2(a1lo.z, b1lo.z, a1hi.z, b1hi.z);
        av[1] = iterm2(a1lo.w, b1lo.w, a1hi.w, b1hi.w);
        c1 = __builtin_amdgcn_wmma_f32_16x16x4_f32(false, av, false, bones,
                                                   (short)0, c1, false, false);
    }

    // C/D layout: lanes 0-15 hold rows 0-7 (vgpr r -> M=r), lanes 16-31 rows 8-15.
    // Lanes 0 and 16 stage their 8 row sums (all columns equal) to LDS.
    if (mrow == 0) {
        const int rbase = half * 8;
#pragma unroll
        for (int r = 0; r < 8; ++r) {
            s[wave][0][rbase + r] = c0[r];
            s[wave][1][rbase + r] = c1[r];
        }
    }
    __syncthreads();

    if (lane < 16) {
        const int q = pbase + lane;
        if (q < N_PAIRS) {
            const int j1  = idx[2 * q + 1];
            const float lv0 = LN2F * s[wave][0][lane];   // log2 -> ln
            const float lv1 = LN2F * s[wave][1][lane];
            const float w0 = weights[0], w1 = weights[1];
            const float logAB = lse2(w0 + lv0, w1 + lv1) - lse2(w0, w1);
            const float logP  = logAB - unary[j1];
            out_logp[q] = logP;
            out_p[q]    = rexp2(logP * INV_LN2F);
        }
    }
}

// ---------------------------------------------------------------------------
extern "C" void kernel_launch(void* const* d_in, const int* in_sizes, int n_in,
                              void* d_out, int out_size, void* d_ws, size_t ws_size,
                              hipStream_t stream) {
    (void)in_sizes; (void)n_in; (void)out_size; (void)d_ws; (void)ws_size;
    const int*   idx = (const int*)d_in[0];     // box_indices (N_PAIRS, 2) int32
    const float* emb = (const float*)d_in[1];   // box_embeddings_orig (2,100000,2,64) f32
    const float* w   = (const float*)d_in[2];   // weights (2,) f32

    float* out        = (float*)d_out;
    float* out_unary  = out;                        // 100000
    float* out_logp   = out + NUM_BOXES;            // 500000
    float* out_p      = out + NUM_BOXES + N_PAIRS;  // 500000

    unary_kernel<<<NUM_BOXES / 8, 256, 0, stream>>>(emb, w, out_unary);

    const int nblocks = (N_PAIRS + 127) / 128;      // 128 pairs per 256-thread block
    pair_kernel<<<nblocks, 256, 0, stream>>>(idx, emb, w, out_unary, out_logp, out_p);
}